// RGCNBlock_7902739824904
// MI455X (gfx1250) — compile-verified
//
#include <hip/hip_runtime.h>
#include <hip/hip_bf16.h>

#define DIM   128
#define RREL  8
// LDS layout: pair-index p = k/2 (64 pairs), per pair a row of 128 cols * 2 floats,
// padded to 144 col-slots -> stride 288 floats. 288 % 64 == 32, so the two
// half-wave K-slices land on disjoint 32-bank halves; a b64 load uses all 64 banks.
#define LDP   288

typedef float v2f __attribute__((ext_vector_type(2)));
typedef float v8f __attribute__((ext_vector_type(8)));

// ---------------------------------------------------------------------------
// Kernel 1: per-(dst, relation) edge counts
// ---------------------------------------------------------------------------
__global__ __launch_bounds__(256) void rgcn_count_kernel(
    const int* __restrict__ edge_index,   // (2, E)
    const int* __restrict__ edge_type,    // (E,)
    unsigned*  __restrict__ cnt,          // (N*R,)
    int nE)
{
    int e = blockIdx.x * 256 + threadIdx.x;
    if (e >= nE) return;
    int dst = edge_index[nE + e];
    int rt  = edge_type[e];
    atomicAdd(&cnt[(size_t)dst * RREL + rt], 1u);
}

// ---------------------------------------------------------------------------
// Kernel 2: WMMA f32 GEMM.
//   blockIdx.y = r in [0,8]: r<8 -> xw[r] = x @ W_rel[r]
//                            r==8 -> agg   = x @ W_root + bias
//   Block: 128 rows x 128 cols. Wave: 16 rows x 128 cols.
//   V_WMMA_F32_16X16X4_F32, K-loop of 32 steps; B staged K-pair-interleaved in
//   LDS so each B fragment is one conflict-free ds_load_b64.
// ---------------------------------------------------------------------------
__global__ __launch_bounds__(256) void rgcn_gemm_kernel(
    const float* __restrict__ x,
    const float* __restrict__ W_rel,
    const float* __restrict__ W_root,
    const float* __restrict__ bias,
    float* __restrict__ xw,
    float* __restrict__ agg,
    int nN)
{
    __shared__ float Wlds[64 * LDP];      // 73,728 B

    const int r = blockIdx.y;
    const float* __restrict__ W = (r < RREL) ? (W_rel + (size_t)r * DIM * DIM) : W_root;

    // Stage weights, interleaving K-pairs: Wlds[p*LDP + col*2 + {0,1}] = W[2p][col], W[2p+1][col].
    // Each trip: 2 global b128 loads -> 2 ds b128 stores (transpose on store).
    for (int i = threadIdx.x; i < 64 * (DIM / 4); i += 256) {
        int p    = i >> 5;                // pair index 0..63
        int colq = (i & 31) * 4;          // col group of 4
        float4 g0 = *(const float4*)(W + (size_t)(2 * p)     * DIM + colq);
        float4 g1 = *(const float4*)(W + (size_t)(2 * p + 1) * DIM + colq);
        float* d = &Wlds[p * LDP + colq * 2];
        float4 s0 = {g0.x, g1.x, g0.y, g1.y};
        float4 s1 = {g0.z, g1.z, g0.w, g1.w};
        *(float4*)(d + 0) = s0;
        *(float4*)(d + 4) = s1;
    }
    __syncthreads();

    const int wave = threadIdx.x >> 5;
    const int lane = threadIdx.x & 31;
    const int row_base = blockIdx.x * 128 + wave * 16;
    if (row_base + 16 > nN) return;       // nN % 16 == 0: tiles are always full

    const int hl = lane >> 4;             // half-wave selects K pair
    const int ll = lane & 15;
    const float* __restrict__ arow = x + (size_t)(row_base + ll) * DIM;

    v8f acc[8];
    const v8f zero = {0.f, 0.f, 0.f, 0.f, 0.f, 0.f, 0.f, 0.f};
#pragma unroll
    for (int c = 0; c < 8; ++c) acc[c] = zero;

    for (int k0 = 0; k0 < DIM; k0 += 4) {
        const int ka = k0 + 2 * hl;       // lanes 0-15: K={k0,k0+1}; lanes 16-31: K={k0+2,k0+3}
        v2f a;
        a.x = arow[ka];                   // contiguous -> one global_load_b64
        a.y = arow[ka + 1];
        const float* brow = &Wlds[(ka >> 1) * LDP + ll * 2];
#pragma unroll
        for (int c = 0; c < 8; ++c) {
            v2f b = *(const v2f*)(brow + c * 32);   // one aligned ds_load_b64
            acc[c] = __builtin_amdgcn_wmma_f32_16x16x4_f32(
                false, a, false, b, (short)0, acc[c], false, false);
        }
    }

    // D layout: VGPR j -> M = j (lanes 0-15) / j+8 (lanes 16-31), N = ll
    if (r < RREL) {
        float* base = xw + ((size_t)r * nN + row_base) * DIM;
#pragma unroll
        for (int c = 0; c < 8; ++c)
#pragma unroll
            for (int j = 0; j < 8; ++j)
                base[(size_t)(j + 8 * hl) * DIM + c * 16 + ll] = acc[c][j];
    } else {
        float* base = agg + (size_t)row_base * DIM;
#pragma unroll
        for (int c = 0; c < 8; ++c) {
            float bia = bias[c * 16 + ll];
#pragma unroll
            for (int j = 0; j < 8; ++j)
                base[(size_t)(j + 8 * hl) * DIM + c * 16 + ll] = acc[c][j] + bia;
        }
    }
}

// ---------------------------------------------------------------------------
// Kernel 3: one wave per edge. Edge metadata forced wave-uniform (scalar path);
// 512B coalesced gather of xw row, normalize, global_atomic_add_f32 scatter.
// ---------------------------------------------------------------------------
__global__ __launch_bounds__(256) void rgcn_scatter_kernel(
    const int*      __restrict__ edge_index,
    const int*      __restrict__ edge_type,
    const unsigned* __restrict__ cnt,
    const float*    __restrict__ xw,
    float*          __restrict__ agg,
    int nE, int nN)
{
    int wid  = __builtin_amdgcn_readfirstlane((blockIdx.x * 256 + threadIdx.x) >> 5);
    int lane = threadIdx.x & 31;
    if (wid >= nE) return;

    int src = edge_index[wid];
    int dst = edge_index[nE + wid];
    int rt  = edge_type[wid];

    float c    = (float)cnt[(size_t)dst * RREL + rt];
    float norm = 1.0f / fmaxf(c, 1.0f);

    const float4 v = *(const float4*)(xw + ((size_t)rt * nN + src) * DIM + lane * 4);
    float* a = agg + (size_t)dst * DIM + lane * 4;
    atomicAdd(a + 0, v.x * norm);
    atomicAdd(a + 1, v.y * norm);
    atomicAdd(a + 2, v.z * norm);
    atomicAdd(a + 3, v.w * norm);
}

// ---------------------------------------------------------------------------
// Kernel 4: one wave per row. LayerNorm (wave32 shfl_xor reduction) + ReLU.
// ---------------------------------------------------------------------------
__global__ __launch_bounds__(256) void rgcn_ln_kernel(
    const float* __restrict__ agg,
    const float* __restrict__ gamma,
    const float* __restrict__ beta,
    float*       __restrict__ out,
    int nN)
{
    int row  = __builtin_amdgcn_readfirstlane((blockIdx.x * 256 + threadIdx.x) >> 5);
    int lane = threadIdx.x & 31;
    if (row >= nN) return;

    float4 v = *(const float4*)(agg + (size_t)row * DIM + lane * 4);
    float s  = v.x + v.y + v.z + v.w;
    float s2 = v.x * v.x + v.y * v.y + v.z * v.z + v.w * v.w;
#pragma unroll
    for (int m = 16; m >= 1; m >>= 1) {
        s  += __shfl_xor(s,  m, 32);
        s2 += __shfl_xor(s2, m, 32);
    }
    float mu  = s * (1.0f / DIM);
    float var = s2 * (1.0f / DIM) - mu * mu;
    float rs  = rsqrtf(var + 1e-5f);

    float4 g = *(const float4*)(gamma + lane * 4);
    float4 b = *(const float4*)(beta  + lane * 4);
    float4 o;
    o.x = fmaxf((v.x - mu) * rs * g.x + b.x, 0.0f);
    o.y = fmaxf((v.y - mu) * rs * g.y + b.y, 0.0f);
    o.z = fmaxf((v.z - mu) * rs * g.z + b.z, 0.0f);
    o.w = fmaxf((v.w - mu) * rs * g.w + b.w, 0.0f);
    *(float4*)(out + (size_t)row * DIM + lane * 4) = o;
}

// ---------------------------------------------------------------------------
extern "C" void kernel_launch(void* const* d_in, const int* in_sizes, int n_in,
                              void* d_out, int out_size, void* d_ws, size_t ws_size,
                              hipStream_t stream) {
    const float* x      = (const float*)d_in[0];
    const int*   ei     = (const int*)  d_in[1];   // (2, E)
    const int*   et     = (const int*)  d_in[2];   // (E,)
    const float* W_rel  = (const float*)d_in[4];   // (R, D, D)
    const float* W_root = (const float*)d_in[5];   // (D, D)
    const float* bias   = (const float*)d_in[6];
    const float* gamma  = (const float*)d_in[7];
    const float* beta   = (const float*)d_in[8];
    float*       out    = (float*)d_out;

    const int nN = in_sizes[0] / DIM;      // 50000
    const int nE = in_sizes[2];            // 1600000

    // workspace layout: xw (R*N*D f32) | agg (N*D f32) | cnt (N*R u32)
    float*    xw  = (float*)d_ws;
    float*    agg = xw + (size_t)RREL * nN * DIM;
    unsigned* cnt = (unsigned*)(agg + (size_t)nN * DIM);

    hipMemsetAsync(cnt, 0, (size_t)nN * RREL * sizeof(unsigned), stream);

    rgcn_count_kernel<<<(nE + 255) / 256, 256, 0, stream>>>(ei, et, cnt, nE);

    dim3 ggrid((nN + 127) / 128, RREL + 1);
    rgcn_gemm_kernel<<<ggrid, 256, 0, stream>>>(x, W_rel, W_root, bias, xw, agg, nN);

    int scat_blocks = (nE * 32 + 255) / 256;
    rgcn_scatter_kernel<<<scat_blocks, 256, 0, stream>>>(ei, et, cnt, xw, agg, nE, nN);

    int ln_blocks = (nN * 32 + 255) / 256;
    rgcn_ln_kernel<<<ln_blocks, 256, 0, stream>>>(agg, gamma, beta, out, nN);
}